// SpectrumMatchingLoss_51943334477834
// MI455X (gfx1250) — compile-verified
//
#include <hip/hip_runtime.h>

typedef __attribute__((ext_vector_type(2))) float v2f;
typedef __attribute__((ext_vector_type(4))) float v4f;
typedef __attribute__((ext_vector_type(8))) float v8f;

#define BB 128
#define SS 48
#define VV 4096
#define PP 4096
#define NROWS (BB * SS)          // 6144
#define NRES  (SS - 2)           // 46 residues
#define NION  (NRES - 1)         // 45 ions per series
#define NTHEO (3 * NION)         // 135 theoretical peaks

#define PROTON_MASS 1.007276f
#define WATER_MASS  18.010565f
#define CO_MASS     27.994915f

#define KSPLIT 8
#define KCH (VV / KSPLIT)        // 512

// ---------------------------------------------------------------------------
// Kernel 0: zero the accumulation regions of the workspace
// ---------------------------------------------------------------------------
__global__ void zero_ws_kernel(float* __restrict__ ws, int n) {
    int i = blockIdx.x * blockDim.x + threadIdx.x;
    if (i < n) ws[i] = 0.0f;
}

// ---------------------------------------------------------------------------
// Kernel 1: expected[b,s] = dot(sequence_probs[b,s,:], aa_masses) via
// V_WMMA_F32_16X16X4_F32. One wave handles a 16-row M-tile over a 512-deep
// K-partition; partial sums accumulate with global f32 atomics.
//
// Layout per CDNA5 ISA (32-bit A 16x4): lanes 0-15 hold K=0,1 in
// VGPR0/1, lanes 16-31 hold K=2,3. We assign memory columns k..k+3 to the
// lower half-wave and k+4..k+7 to the upper half-wave (dot products are
// order-independent), and build B by broadcasting the matching aa_masses
// values, so every column of D ends up holding the same 16 dot products.
// ---------------------------------------------------------------------------
__global__ __launch_bounds__(256) void gemv_wmma_kernel(
    const float* __restrict__ probs,   // [NROWS, VV]
    const float* __restrict__ aa,      // [VV]
    float* __restrict__ expected)      // [NROWS], pre-zeroed
{
    const int wave  = (blockIdx.x * blockDim.x + threadIdx.x) >> 5;
    const int lane  = threadIdx.x & 31;
    const int tile  = wave >> 3;        // 0..383
    const int kpart = wave & (KSPLIT - 1);
    if (tile >= NROWS / 16) return;     // wave-uniform; EXEC stays all-ones

    const int half = lane >> 4;         // 0: K k..k+3, 1: K k+4..k+7
    const int mrow = lane & 15;
    const long long rowbase = (long long)(tile * 16 + mrow) * VV;
    const int k0 = kpart * KCH;

    const float* ap = probs + rowbase + k0 + 4 * half;
    const float* bp = aa + k0 + 4 * half;

    v8f c = {};
#pragma unroll 4
    for (int k = 0; k < KCH; k += 8) {
        v4f a = *(const v4f*)(ap + k);  // 16B-aligned b128 load
        v4f b = *(const v4f*)(bp + k);  // L0-resident (aa is 16 KB)
        v2f a0 = {a.x, a.y}, b0 = {b.x, b.y};
        v2f a1 = {a.z, a.w}, b1 = {b.z, b.w};
        c = __builtin_amdgcn_wmma_f32_16x16x4_f32(
                false, a0, false, b0, (short)0, c, false, false);
        c = __builtin_amdgcn_wmma_f32_16x16x4_f32(
                false, a1, false, b1, (short)0, c, false, false);
    }

    // D layout: VGPR j, lanes 0-15 -> row j; lanes 16-31 -> row j+8.
    // All N columns are identical, so lanes 0 and 16 publish the 16 rows.
    if ((lane & 15) == 0) {
        const int base = tile * 16 + half * 8;
#pragma unroll
        for (int j = 0; j < 8; ++j)
            __hip_atomic_fetch_add(&expected[base + j], c[j],
                                   __ATOMIC_RELAXED, __HIP_MEMORY_SCOPE_AGENT);
    }
}

// ---------------------------------------------------------------------------
// Kernel 2: per-batch soft spectrum matching loss. One block per batch.
// ---------------------------------------------------------------------------
__global__ __launch_bounds__(256) void loss_kernel(
    const float* __restrict__ expected,     // [NROWS]
    const float* __restrict__ obs_mass,     // [BB, PP]
    const float* __restrict__ obs_inten,    // [BB, PP]
    const unsigned char* __restrict__ pmask,// [BB, PP] (bool)
    float* __restrict__ acc)                // [2]: {numerator, denominator}
{
    __shared__ float2 peaks[PP];            // (mass-or-+inf, intensity): 32 KB
    __shared__ float  theo[NTHEO];
    __shared__ float  red[16];

    const int b   = blockIdx.x;
    const int tid = threadIdx.x;

    // Stage observed peaks; fold peak_mask into the mass (masked -> far away)
    for (int p = tid; p < PP; p += blockDim.x) {
        const int idx = b * PP + p;
        const float m = obs_mass[idx];
        const float w = obs_inten[idx];
        peaks[p] = make_float2(pmask[idx] ? m : 3.0e30f, w);
    }

    // Theoretical b/y/a ions from expected residue masses (46 residues)
    if (tid == 0) {
        float resid[NRES];
        float total = 0.0f;
        for (int i = 0; i < NRES; ++i) {
            resid[i] = expected[b * SS + 1 + i];
            total += resid[i];
        }
        float run = 0.0f;
        for (int i = 0; i < NION; ++i) {
            theo[NION + i] = total - run + (WATER_MASS + PROTON_MASS); // y
            run += resid[i];
            const float bi = run + PROTON_MASS;                        // b
            theo[i] = bi;
            theo[2 * NION + i] = bi - CO_MASS;                         // a
        }
    }
    __syncthreads();

    const int wave = tid >> 5;
    const int lane = tid & 31;
    float num = 0.0f, den = 0.0f;

    for (int t = wave; t < NTHEO; t += 8) {
        const float tv = theo[t];
        float se = 0.0f, wh = 0.0f, wi = 0.0f;
        for (int p = lane; p < PP; p += 32) {
            const float2 pk = peaks[p];
            const float d = fabsf(tv - pk.x);
            if (d < 0.5f) {                       // in window & unmasked
                const float e = __expf(-10.0f * d);   // score = -d/0.1
                const float hub = (d <= 0.2f) ? 0.5f * d * d
                                              : 0.2f * (d - 0.1f);
                se += e; wh += e * hub; wi += e * pk.y;
            }
        }
#pragma unroll
        for (int off = 16; off > 0; off >>= 1) {
            se += __shfl_xor(se, off, 32);
            wh += __shfl_xor(wh, off, 32);
            wi += __shfl_xor(wi, off, 32);
        }
        if (lane == 0 && se > 0.0f) {
            num += (wh / se) * (wi / se);   // matched_dist * intensity_weight
            den += 1.0f;                    // num_matches
        }
    }

    if (lane == 0) { red[2 * wave] = num; red[2 * wave + 1] = den; }
    __syncthreads();
    if (tid == 0) {
        float n = 0.0f, d = 0.0f;
        for (int w = 0; w < 8; ++w) { n += red[2 * w]; d += red[2 * w + 1]; }
        __hip_atomic_fetch_add(&acc[0], n, __ATOMIC_RELAXED,
                               __HIP_MEMORY_SCOPE_AGENT);
        __hip_atomic_fetch_add(&acc[1], d, __ATOMIC_RELAXED,
                               __HIP_MEMORY_SCOPE_AGENT);
    }
}

// ---------------------------------------------------------------------------
// Kernel 3: finalize scalar loss
// ---------------------------------------------------------------------------
__global__ void finalize_kernel(const float* __restrict__ acc,
                                float* __restrict__ out) {
    if (blockIdx.x == 0 && threadIdx.x == 0)
        out[0] = acc[0] / fmaxf(acc[1], 1.0f);
}

// ---------------------------------------------------------------------------
extern "C" void kernel_launch(void* const* d_in, const int* in_sizes, int n_in,
                              void* d_out, int out_size, void* d_ws,
                              size_t ws_size, hipStream_t stream) {
    const float* probs = (const float*)d_in[0];          // [B,S,V] f32
    const float* omass = (const float*)d_in[1];          // [B,P] f32
    const float* ointe = (const float*)d_in[2];          // [B,P] f32
    const float* aa    = (const float*)d_in[3];          // [V] f32
    const unsigned char* pmask = (const unsigned char*)d_in[4]; // [B,P] bool

    float* ws       = (float*)d_ws;
    float* expected = ws;              // NROWS floats
    float* acc      = ws + NROWS;      // 2 floats

    const int nz = NROWS + 2;
    zero_ws_kernel<<<(nz + 255) / 256, 256, 0, stream>>>(ws, nz);

    // 384 M-tiles x 8 K-partitions = 3072 waves = 384 blocks of 8 waves
    const int nwaves = (NROWS / 16) * KSPLIT;
    gemv_wmma_kernel<<<nwaves / 8, 256, 0, stream>>>(probs, aa, expected);

    loss_kernel<<<BB, 256, 0, stream>>>(expected, omass, ointe, pmask, acc);

    finalize_kernel<<<1, 64, 0, stream>>>(acc, (float*)d_out);
}